// Graphsage_22754736734507
// MI455X (gfx1250) — compile-verified
//
#include <hip/hip_runtime.h>
#include <hip/hip_bf16.h>
#include <math.h>

// GraphSAGE 2-layer forward for MI455X (gfx1250), f32 WMMA path.
//
// d_in order: x[100000*64], edge_index[2*E] (int), W1_l[64*128], W1_r[64*128],
//             b1[128], W2_l[128*40], W2_r[128*40], b2[40]
// d_out: log_softmax logits [n*40] f32.

typedef __attribute__((ext_vector_type(2))) float v2f;
typedef __attribute__((ext_vector_type(8))) float v8f;

#define D_IN  64
#define D_HID 128
#define NCLS  40

// ---------------------------------------------------------------- zero fill
__global__ void sage_zero(float* __restrict__ p, long long count) {
    long long i = (long long)blockIdx.x * blockDim.x + threadIdx.x;
    long long stride = (long long)gridDim.x * blockDim.x;
    for (; i < count; i += stride) p[i] = 0.0f;
}

// --------------------------------------------- edge scatter (mean numerator)
// thread t -> edge e = t>>fshift, feature f = t & (feat-1).
// Consecutive lanes share an edge -> coalesced gather of x[src] row and
// coalesced global_atomic_add_f32 into agg[dst] row (L2-resident RMW).
__global__ void sage_scatter(const float* __restrict__ feat_in,
                             const int* __restrict__ esrc,
                             const int* __restrict__ edst,
                             float* __restrict__ agg,
                             float* __restrict__ deg,
                             long long total, int fshift, int do_deg) {
    long long t = (long long)blockIdx.x * blockDim.x + threadIdx.x;
    if (t >= total) return;
    int feat = 1 << fshift;
    int e = (int)(t >> fshift);
    int f = (int)(t & (feat - 1));
    int s = esrc[e];
    int d = edst[e];
    atomicAdd(&agg[(size_t)d * feat + f], feat_in[(size_t)s * feat + f]);
    if (do_deg && f == 0) atomicAdd(&deg[d], 1.0f);
}

// ------------------------------------------------------------ layer-1 GEMM
// h1 = relu( (agg1/deg) @ W1_l  +  x @ W1_r  + b1 )    [n x 128]
// 256 threads = 8 waves; block owns 16 rows; wave w owns N-tile w (16 cols).
// K=64 -> 16 k-steps of V_WMMA_F32_16X16X4_F32 per product (32 WMMAs/tile).
__global__ __launch_bounds__(256) void sage_gemm1(
    const float* __restrict__ x, const float* __restrict__ agg,
    const float* __restrict__ deg, const float* __restrict__ Wl,
    const float* __restrict__ Wr, const float* __restrict__ bias,
    float* __restrict__ h, int n) {
    __shared__ float s_x[16 * D_IN];
    __shared__ float s_m[16 * D_IN];
    const int row0 = blockIdx.x * 16;
    const int tid  = threadIdx.x;

    // Stage 16 rows of x and of mean = agg/max(deg,1) into LDS.
    for (int i = tid; i < 16 * D_IN; i += 256) {
        int r = i >> 6, f = i & (D_IN - 1);
        int gr = row0 + r;
        float xv = 0.0f, mv = 0.0f;
        if (gr < n) {
            xv = x[(size_t)gr * D_IN + f];
            mv = agg[(size_t)gr * D_IN + f] / fmaxf(deg[gr], 1.0f);
        }
        s_x[i] = xv;
        s_m[i] = mv;
    }
    __syncthreads();

    const int wave  = tid >> 5;          // 0..7 -> N tile
    const int lane  = tid & 31;
    const int col0  = wave * 16;
    const int r     = lane & 15;         // A row within tile / B,C column lane
    const int khalf = (lane >> 4) * 2;   // A/B K sub-block: lanes16-31 -> K+2,K+3
    const int col   = col0 + r;

    v8f c = {};
#pragma unroll
    for (int kk = 0; kk < D_IN / 4; ++kk) {
        int kb = kk * 4 + khalf;
        // mean @ W1_l
        v2f am = { s_m[r * D_IN + kb], s_m[r * D_IN + kb + 1] };
        v2f bl = { Wl[(size_t)kb * D_HID + col], Wl[(size_t)(kb + 1) * D_HID + col] };
        c = __builtin_amdgcn_wmma_f32_16x16x4_f32(false, am, false, bl,
                                                  (short)0, c, false, false);
        // x @ W1_r
        v2f ax = { s_x[r * D_IN + kb], s_x[r * D_IN + kb + 1] };
        v2f br = { Wr[(size_t)kb * D_HID + col], Wr[(size_t)(kb + 1) * D_HID + col] };
        c = __builtin_amdgcn_wmma_f32_16x16x4_f32(false, ax, false, br,
                                                  (short)0, c, false, false);
    }

    // C/D layout: VGPR i -> row (row0+i) for lanes 0-15, (row0+8+i) for 16-31.
    float bv = bias[col];
#pragma unroll
    for (int i = 0; i < 8; ++i) {
        int gr = row0 + i + (lane >> 4) * 8;
        if (gr < n) {
            float v = c[i] + bv;
            h[(size_t)gr * D_HID + col] = fmaxf(v, 0.0f);
        }
    }
}

// ------------------------------------------------------------ layer-2 GEMM
// logits = (agg2/deg) @ W2_l + h1 @ W2_r + b2     [n x 40], N padded to 48.
// 96 threads = 3 waves -> 3 N-tiles of 16 (cols >= 40 masked).
__global__ __launch_bounds__(96) void sage_gemm2(
    const float* __restrict__ h, const float* __restrict__ agg,
    const float* __restrict__ deg, const float* __restrict__ Wl,
    const float* __restrict__ Wr, const float* __restrict__ bias,
    float* __restrict__ out, int n) {
    __shared__ float s_h[16 * D_HID];
    __shared__ float s_m[16 * D_HID];
    const int row0 = blockIdx.x * 16;
    const int tid  = threadIdx.x;

    for (int i = tid; i < 16 * D_HID; i += 96) {
        int r = i >> 7, f = i & (D_HID - 1);
        int gr = row0 + r;
        float hv = 0.0f, mv = 0.0f;
        if (gr < n) {
            hv = h[(size_t)gr * D_HID + f];
            mv = agg[(size_t)gr * D_HID + f] / fmaxf(deg[gr], 1.0f);
        }
        s_h[i] = hv;
        s_m[i] = mv;
    }
    __syncthreads();

    const int wave  = tid >> 5;          // 0..2
    const int lane  = tid & 31;
    const int col0  = wave * 16;
    const int r     = lane & 15;
    const int khalf = (lane >> 4) * 2;
    const int col   = col0 + r;
    const bool colok = (col < NCLS);

    v8f c = {};
#pragma unroll
    for (int kk = 0; kk < D_HID / 4; ++kk) {
        int kb = kk * 4 + khalf;
        v2f am = { s_m[r * D_HID + kb], s_m[r * D_HID + kb + 1] };
        v2f bl = { colok ? Wl[(size_t)kb * NCLS + col] : 0.0f,
                   colok ? Wl[(size_t)(kb + 1) * NCLS + col] : 0.0f };
        c = __builtin_amdgcn_wmma_f32_16x16x4_f32(false, am, false, bl,
                                                  (short)0, c, false, false);
        v2f ah = { s_h[r * D_HID + kb], s_h[r * D_HID + kb + 1] };
        v2f br = { colok ? Wr[(size_t)kb * NCLS + col] : 0.0f,
                   colok ? Wr[(size_t)(kb + 1) * NCLS + col] : 0.0f };
        c = __builtin_amdgcn_wmma_f32_16x16x4_f32(false, ah, false, br,
                                                  (short)0, c, false, false);
    }

    float bv = colok ? bias[col] : 0.0f;
#pragma unroll
    for (int i = 0; i < 8; ++i) {
        int gr = row0 + i + (lane >> 4) * 8;
        if (gr < n && colok) {
            out[(size_t)gr * NCLS + col] = c[i] + bv;
        }
    }
}

// -------------------------------------------------- row-wise log_softmax
__global__ void sage_log_softmax(float* __restrict__ out, int n) {
    int row = blockIdx.x * blockDim.x + threadIdx.x;
    if (row >= n) return;
    float* p = out + (size_t)row * NCLS;
    float m = -INFINITY;
#pragma unroll
    for (int j = 0; j < NCLS; ++j) m = fmaxf(m, p[j]);
    float s = 0.0f;
#pragma unroll
    for (int j = 0; j < NCLS; ++j) s += __expf(p[j] - m);
    float ls = __logf(s);
#pragma unroll
    for (int j = 0; j < NCLS; ++j) p[j] = p[j] - m - ls;
}

// ----------------------------------------------------------------- launcher
extern "C" void kernel_launch(void* const* d_in, const int* in_sizes, int n_in,
                              void* d_out, int out_size, void* d_ws, size_t ws_size,
                              hipStream_t stream) {
    const float* x    = (const float*)d_in[0];
    const int*   eidx = (const int*)d_in[1];
    const float* W1l  = (const float*)d_in[2];
    const float* W1r  = (const float*)d_in[3];
    const float* b1   = (const float*)d_in[4];
    const float* W2l  = (const float*)d_in[5];
    const float* W2r  = (const float*)d_in[6];
    const float* b2   = (const float*)d_in[7];
    float* out = (float*)d_out;

    const int n  = in_sizes[0] / D_IN;       // 100000 nodes
    const int nE = in_sizes[1] / 2;          // 1,000,000 edges
    const int* esrc = eidx;
    const int* edst = eidx + nE;

    // Workspace layout (all f32): deg[n] | agg1[n*64] | h1[n*128] | agg2[n*128]
    float* deg  = (float*)d_ws;
    float* agg1 = deg  + (size_t)n;
    float* h1   = agg1 + (size_t)n * D_IN;
    float* agg2 = h1   + (size_t)n * D_HID;

    const int mtiles = (n + 15) / 16;

    // 1) Zero the atomic accumulators (deg+agg1 contiguous, then agg2).
    long long z1 = (long long)n * (1 + D_IN);
    long long z2 = (long long)n * D_HID;
    sage_zero<<<4096, 256, 0, stream>>>(deg, z1);
    sage_zero<<<4096, 256, 0, stream>>>(agg2, z2);

    // 2) Layer-1 edge aggregation (+ degree counts).
    {
        long long total = (long long)nE * D_IN;
        long long blocks = (total + 255) / 256;
        sage_scatter<<<(unsigned)blocks, 256, 0, stream>>>(
            x, esrc, edst, agg1, deg, total, 6 /*log2(64)*/, 1);
    }

    // 3) h1 = relu(mean1 @ W1_l + x @ W1_r + b1)   -- WMMA f32
    sage_gemm1<<<mtiles, 256, 0, stream>>>(x, agg1, deg, W1l, W1r, b1, h1, n);

    // 4) Layer-2 edge aggregation over h1.
    {
        long long total = (long long)nE * D_HID;
        long long blocks = (total + 255) / 256;
        sage_scatter<<<(unsigned)blocks, 256, 0, stream>>>(
            h1, esrc, edst, agg2, deg, total, 7 /*log2(128)*/, 0);
    }

    // 5) logits = mean2 @ W2_l + h1 @ W2_r + b2 -> d_out  -- WMMA f32
    sage_gemm2<<<mtiles, 96, 0, stream>>>(h1, agg2, deg, W2l, W2r, b2, out, n);

    // 6) In-place log_softmax over the 40 classes.
    sage_log_softmax<<<(n + 255) / 256, 256, 0, stream>>>(out, n);
}